// HAN_38104949850441
// MI455X (gfx1250) — compile-verified
//
#include <hip/hip_runtime.h>
#include <hip/hip_bf16.h>
#include <math.h>

#define NU 50000
#define NI 50000
#define FIN 128
#define HIDF 256
#define NHEAD 8
#define DHEAD 32
#define OUTF 64
#define NEDGE 800000
#define NEGS 0.2f

typedef float v2f __attribute__((ext_vector_type(2)));
typedef float v8f __attribute__((ext_vector_type(8)));

// ---- order-preserving float <-> uint key for atomic max ----
__device__ __forceinline__ unsigned fkey(float f) {
  unsigned b = __float_as_uint(f);
  return (b & 0x80000000u) ? ~b : (b | 0x80000000u);
}
__device__ __forceinline__ float funkey(unsigned k) {
  return __uint_as_float((k & 0x80000000u) ? (k ^ 0x80000000u) : ~k);
}

__device__ __forceinline__ v2f load_b2(const float* b0p, const float* b1p, int o) {
  v2f b; b[0] = b0p[o]; b[1] = b1p[o]; return b;
}

#define WMMA_F32(A_, B_, C_) \
  __builtin_amdgcn_wmma_f32_16x16x4_f32(false, (A_), false, (B_), (short)0, (C_), false, false)

// =====================================================================
// WMMA GEMM: C[M,N] = A[M,K] @ B[K,N] + bias[N]   (all row-major, f32)
// one wave -> 16 rows x 64 cols; software-pipelined operand prefetch.
// =====================================================================
__global__ void wmma_gemm_bias(const float* __restrict__ A, const float* __restrict__ B,
                               const float* __restrict__ bias, float* __restrict__ C,
                               int M, int K, int N) {
  const int lane = threadIdx.x & 31;
  const int gw   = blockIdx.x * (blockDim.x >> 5) + (threadIdx.x >> 5);
  const int rowTiles = M >> 4, colGroups = N >> 6;
  if (gw >= rowTiles * colGroups) return;
  const int rt = gw % rowTiles, cg = gw / rowTiles;
  const int row  = (rt << 4) + (lane & 15);
  const int col0 = cg << 6;
  const int koff = (lane >> 4) << 1;
  v8f c0 = {0,0,0,0,0,0,0,0}, c1 = c0, c2 = c0, c3 = c0;
  const float* arow  = A + (size_t)row * K;
  const float* bbase = B + col0 + (lane & 15);

  v2f a; a[0] = arow[koff]; a[1] = arow[koff + 1];
  const float* b0p = bbase + (size_t)koff * N;
  const float* b1p = b0p + N;
  v2f b0 = load_b2(b0p, b1p, 0),  b1 = load_b2(b0p, b1p, 16);
  v2f b2 = load_b2(b0p, b1p, 32), b3 = load_b2(b0p, b1p, 48);

  for (int k = 4; k < K; k += 4) {
    v2f an; an[0] = arow[k + koff]; an[1] = arow[k + koff + 1];
    const float* nb0 = bbase + (size_t)(k + koff) * N;
    const float* nb1 = nb0 + N;
    v2f n0 = load_b2(nb0, nb1, 0),  n1 = load_b2(nb0, nb1, 16);
    v2f n2 = load_b2(nb0, nb1, 32), n3 = load_b2(nb0, nb1, 48);
    c0 = WMMA_F32(a, b0, c0); c1 = WMMA_F32(a, b1, c1);
    c2 = WMMA_F32(a, b2, c2); c3 = WMMA_F32(a, b3, c3);
    a = an; b0 = n0; b1 = n1; b2 = n2; b3 = n3;
  }
  c0 = WMMA_F32(a, b0, c0); c1 = WMMA_F32(a, b1, c1);
  c2 = WMMA_F32(a, b2, c2); c3 = WMMA_F32(a, b3, c3);

  const int rbase = (lane >> 4) << 3;
  const int cc = col0 + (lane & 15);
  for (int r = 0; r < 8; ++r) {
    int rr = (rt << 4) + rbase + r;
    float* cp = C + (size_t)rr * N;
    cp[cc]      = c0[r] + bias[cc];
    cp[cc + 16] = c1[r] + bias[cc + 16];
    cp[cc + 32] = c2[r] + bias[cc + 32];
    cp[cc + 48] = c3[r] + bias[cc + 48];
  }
}

// =====================================================================
// Semantic-attention score GEMM: colsum[j] += sum_rows tanh( (relu(Z)@Wk + bk)[.,j] )
// output never stored, only column sums.
// =====================================================================
__global__ void wmma_tanh_colsum(const float* __restrict__ Z, const float* __restrict__ Wkm,
                                 const float* __restrict__ bkv, float* __restrict__ colsum, int M) {
  const int lane = threadIdx.x & 31;
  const int gw   = blockIdx.x * (blockDim.x >> 5) + (threadIdx.x >> 5);
  const int rowTiles = M >> 4, colGroups = HIDF >> 6;
  if (gw >= rowTiles * colGroups) return;
  const int rt = gw % rowTiles, cg = gw / rowTiles;
  const int row  = (rt << 4) + (lane & 15);
  const int col0 = cg << 6;
  const int koff = (lane >> 4) << 1;
  v8f c0 = {0,0,0,0,0,0,0,0}, c1 = c0, c2 = c0, c3 = c0;
  const float* zrow  = Z + (size_t)row * HIDF;
  const float* bbase = Wkm + col0 + (lane & 15);

  v2f a; a[0] = fmaxf(zrow[koff], 0.f); a[1] = fmaxf(zrow[koff + 1], 0.f);
  const float* b0p = bbase + (size_t)koff * HIDF;
  const float* b1p = b0p + HIDF;
  v2f b0 = load_b2(b0p, b1p, 0),  b1 = load_b2(b0p, b1p, 16);
  v2f b2 = load_b2(b0p, b1p, 32), b3 = load_b2(b0p, b1p, 48);

  for (int k = 4; k < HIDF; k += 4) {
    v2f an; an[0] = fmaxf(zrow[k + koff], 0.f); an[1] = fmaxf(zrow[k + koff + 1], 0.f);
    const float* nb0 = bbase + (size_t)(k + koff) * HIDF;
    const float* nb1 = nb0 + HIDF;
    v2f n0 = load_b2(nb0, nb1, 0),  n1 = load_b2(nb0, nb1, 16);
    v2f n2 = load_b2(nb0, nb1, 32), n3 = load_b2(nb0, nb1, 48);
    c0 = WMMA_F32(a, b0, c0); c1 = WMMA_F32(a, b1, c1);
    c2 = WMMA_F32(a, b2, c2); c3 = WMMA_F32(a, b3, c3);
    a = an; b0 = n0; b1 = n1; b2 = n2; b3 = n3;
  }
  c0 = WMMA_F32(a, b0, c0); c1 = WMMA_F32(a, b1, c1);
  c2 = WMMA_F32(a, b2, c2); c3 = WMMA_F32(a, b3, c3);

  const int cc = col0 + (lane & 15);
  float bk0 = bkv[cc], bk1 = bkv[cc + 16], bk2 = bkv[cc + 32], bk3 = bkv[cc + 48];
  float s0 = 0.f, s1 = 0.f, s2 = 0.f, s3 = 0.f;
  for (int r = 0; r < 8; ++r) {
    s0 += tanhf(c0[r] + bk0); s1 += tanhf(c1[r] + bk1);
    s2 += tanhf(c2[r] + bk2); s3 += tanhf(c3[r] + bk3);
  }
  // lanes l and l^16 hold the same columns (different row halves) -> merge
  s0 += __shfl_xor(s0, 16); s1 += __shfl_xor(s1, 16);
  s2 += __shfl_xor(s2, 16); s3 += __shfl_xor(s3, 16);
  if (lane < 16) {
    atomicAdd(&colsum[cc],      s0);
    atomicAdd(&colsum[cc + 16], s1);
    atomicAdd(&colsum[cc + 32], s2);
    atomicAdd(&colsum[cc + 48], s3);
  }
}

// =====================================================================
// Final: out = (attn0*relu(ZA) + attn1*relu(ZB)) @ Wl + bl    N=64, K=256
// =====================================================================
__global__ void wmma_final(const float* __restrict__ ZA, const float* __restrict__ ZB,
                           const float* __restrict__ attn, const float* __restrict__ Wl,
                           const float* __restrict__ blv, float* __restrict__ out, int M) {
  const int lane = threadIdx.x & 31;
  const int gw   = blockIdx.x * (blockDim.x >> 5) + (threadIdx.x >> 5);
  const int rowTiles = M >> 4;
  if (gw >= rowTiles) return;
  const float a0w = attn[0], a1w = attn[1];
  const int row  = (gw << 4) + (lane & 15);
  const int koff = (lane >> 4) << 1;
  v8f c0 = {0,0,0,0,0,0,0,0}, c1 = c0, c2 = c0, c3 = c0;
  const float* za = ZA + (size_t)row * HIDF;
  const float* zb = ZB + (size_t)row * HIDF;
  const float* bbase = Wl + (lane & 15);

  v2f a;
  a[0] = a0w * fmaxf(za[koff], 0.f)     + a1w * fmaxf(zb[koff], 0.f);
  a[1] = a0w * fmaxf(za[koff + 1], 0.f) + a1w * fmaxf(zb[koff + 1], 0.f);
  const float* b0p = bbase + (size_t)koff * OUTF;
  const float* b1p = b0p + OUTF;
  v2f b0 = load_b2(b0p, b1p, 0),  b1 = load_b2(b0p, b1p, 16);
  v2f b2 = load_b2(b0p, b1p, 32), b3 = load_b2(b0p, b1p, 48);

  for (int k = 4; k < HIDF; k += 4) {
    v2f an;
    an[0] = a0w * fmaxf(za[k + koff], 0.f)     + a1w * fmaxf(zb[k + koff], 0.f);
    an[1] = a0w * fmaxf(za[k + koff + 1], 0.f) + a1w * fmaxf(zb[k + koff + 1], 0.f);
    const float* nb0 = bbase + (size_t)(k + koff) * OUTF;
    const float* nb1 = nb0 + OUTF;
    v2f n0 = load_b2(nb0, nb1, 0),  n1 = load_b2(nb0, nb1, 16);
    v2f n2 = load_b2(nb0, nb1, 32), n3 = load_b2(nb0, nb1, 48);
    c0 = WMMA_F32(a, b0, c0); c1 = WMMA_F32(a, b1, c1);
    c2 = WMMA_F32(a, b2, c2); c3 = WMMA_F32(a, b3, c3);
    a = an; b0 = n0; b1 = n1; b2 = n2; b3 = n3;
  }
  c0 = WMMA_F32(a, b0, c0); c1 = WMMA_F32(a, b1, c1);
  c2 = WMMA_F32(a, b2, c2); c3 = WMMA_F32(a, b3, c3);

  const int rbase = (lane >> 4) << 3;
  const int cc = lane & 15;
  for (int r = 0; r < 8; ++r) {
    int rr = (gw << 4) + rbase + r;
    float* op = out + (size_t)rr * OUTF;
    op[cc]      = c0[r] + blv[cc];
    op[cc + 16] = c1[r] + blv[cc + 16];
    op[cc + 32] = c2[r] + blv[cc + 32];
    op[cc + 48] = c3[r] + blv[cc + 48];
  }
}

// =====================================================================
// Per-node attention logits: o_i[n,h] = dot(h[n, h*32 : h*32+32], a_i[h,:])
// =====================================================================
__global__ void node_logits(const float* __restrict__ hmat,
                            const float* __restrict__ a0v, const float* __restrict__ a1v,
                            const float* __restrict__ a2v, const float* __restrict__ a3v,
                            float* __restrict__ o0, float* __restrict__ o1,
                            float* __restrict__ o2, float* __restrict__ o3, int n) {
  __shared__ float sa[4][HIDF];
  int t = threadIdx.x;                 // blockDim.x == 256 == HIDF
  sa[0][t] = a0v[t]; sa[1][t] = a1v[t]; sa[2][t] = a2v[t]; sa[3][t] = a3v[t];
  __syncthreads();
  int idx = blockIdx.x * blockDim.x + t;
  if (idx >= n * NHEAD) return;
  int node = idx >> 3, h = idx & 7;
  const float* row = hmat + (size_t)node * HIDF + h * DHEAD;
  const float* w0 = &sa[0][h * DHEAD]; const float* w1 = &sa[1][h * DHEAD];
  const float* w2 = &sa[2][h * DHEAD]; const float* w3 = &sa[3][h * DHEAD];
  float s0 = 0.f, s1 = 0.f, s2 = 0.f, s3 = 0.f;
  for (int k = 0; k < DHEAD; ++k) {
    float x = row[k];
    s0 += x * w0[k]; s1 += x * w1[k]; s2 += x * w2[k]; s3 += x * w3[k];
  }
  o0[idx] = s0; o1[idx] = s1; o2[idx] = s2; o3[idx] = s3;
}

// =====================================================================
// Edge-relation pipeline
// =====================================================================
__global__ void rel_init(unsigned* __restrict__ amaxk, float* __restrict__ denom,
                         float* __restrict__ z, int ndst) {
  int i = blockIdx.x * blockDim.x + threadIdx.x;
  if (i < ndst * HIDF) z[i] = 0.f;
  if (i < ndst * NHEAD) { amaxk[i] = 0u; denom[i] = 0.f; }
}

__global__ void rel_logits(const int* __restrict__ src, const int* __restrict__ dst,
                           const float* __restrict__ alsrc, const float* __restrict__ aldst,
                           float* __restrict__ eal, unsigned* __restrict__ amaxk, int ne) {
  int idx = blockIdx.x * blockDim.x + threadIdx.x;
  if (idx >= ne * NHEAD) return;
  int e = idx >> 3, h = idx & 7;
  int s = src[e], d = dst[e];
  float v = alsrc[s * NHEAD + h] + aldst[d * NHEAD + h];
  v = (v > 0.f) ? v : NEGS * v;               // leaky_relu
  eal[idx] = v;
  atomicMax(&amaxk[d * NHEAD + h], fkey(v));  // segment max (ordered-key u32 max)
}

__global__ void rel_exp(const int* __restrict__ dst, float* __restrict__ eal,
                        const unsigned* __restrict__ amaxk, float* __restrict__ denom, int ne) {
  int idx = blockIdx.x * blockDim.x + threadIdx.x;
  if (idx >= ne * NHEAD) return;
  int e = idx >> 3, h = idx & 7;
  int d = dst[e];
  float m  = funkey(amaxk[d * NHEAD + h]);
  float ex = expf(eal[idx] - m);
  eal[idx] = ex;
  atomicAdd(&denom[d * NHEAD + h], ex);       // segment sum
}

// 64 lanes per edge, float4 per lane: coalesced 1KB gather + 4-wide atomic scatter
__global__ void rel_agg(const int* __restrict__ src, const int* __restrict__ dst,
                        const float* __restrict__ xs, const float* __restrict__ eal,
                        const float* __restrict__ denom, float* __restrict__ z, int ne) {
  int t = threadIdx.x;
  int e = blockIdx.x * 4 + (t >> 6);
  if (e >= ne) return;
  int c4 = (t & 63) << 2;                     // element 0..255, step 4 (within one head)
  int h  = c4 >> 5;
  int s = src[e], d = dst[e];
  float alpha = eal[e * NHEAD + h] / (denom[d * NHEAD + h] + 1e-16f);
  const float4 xv = *(const float4*)(xs + (size_t)s * HIDF + c4);
  float* zp = z + (size_t)d * HIDF + c4;
  atomicAdd(zp + 0, xv.x * alpha);
  atomicAdd(zp + 1, xv.y * alpha);
  atomicAdd(zp + 2, xv.z * alpha);
  atomicAdd(zp + 3, xv.w * alpha);
}

__global__ void zero_f32(float* __restrict__ p, int n) {
  int i = blockIdx.x * blockDim.x + threadIdx.x;
  if (i < n) p[i] = 0.f;
}

// score_m = sum_j q[j] * colsum_m[j] / n ;  attn = softmax over the 2 metapaths
__global__ void sem_softmax(const float* __restrict__ csA, const float* __restrict__ csB,
                            const float* __restrict__ q, float* __restrict__ attn, float invN) {
  __shared__ float rA[HIDF], rB[HIDF];
  int t = threadIdx.x;                        // blockDim.x == 256
  float qv = q[t];
  rA[t] = qv * csA[t] * invN;
  rB[t] = qv * csB[t] * invN;
  __syncthreads();
  for (int s = 128; s > 0; s >>= 1) {
    if (t < s) { rA[t] += rA[t + s]; rB[t] += rB[t + s]; }
    __syncthreads();
  }
  if (t == 0) {
    float sA = rA[0], sB = rB[0];
    float m  = fmaxf(sA, sB);
    float eA = expf(sA - m), eB = expf(sB - m);
    float dsum = eA + eB;
    attn[0] = eA / dsum; attn[1] = eB / dsum;
  }
}

// =====================================================================
extern "C" void kernel_launch(void* const* d_in, const int* in_sizes, int n_in,
                              void* d_out, int out_size, void* d_ws, size_t ws_size,
                              hipStream_t stream) {
  (void)in_sizes; (void)n_in; (void)out_size; (void)ws_size;
  const float* x_user  = (const float*)d_in[0];
  const float* x_item  = (const float*)d_in[1];
  const int*   edge_ui = (const int*)d_in[2];
  const int*   edge_iu = (const int*)d_in[3];
  const int*   edge_uu = (const int*)d_in[4];
  const int*   edge_ii = (const int*)d_in[5];
  const float* Wp_user = (const float*)d_in[6];
  const float* bp_user = (const float*)d_in[7];
  const float* Wp_item = (const float*)d_in[8];
  const float* bp_item = (const float*)d_in[9];
  const float* as_ui = (const float*)d_in[10];
  const float* ad_ui = (const float*)d_in[11];
  const float* as_iu = (const float*)d_in[12];
  const float* ad_iu = (const float*)d_in[13];
  const float* as_uu = (const float*)d_in[14];
  const float* ad_uu = (const float*)d_in[15];
  const float* as_ii = (const float*)d_in[16];
  const float* ad_ii = (const float*)d_in[17];
  const float* Wk = (const float*)d_in[18];
  const float* bk = (const float*)d_in[19];
  const float* q  = (const float*)d_in[20];
  const float* Wl = (const float*)d_in[21];
  const float* bl = (const float*)d_in[22];

  // ---- workspace carve-out ----
  char* ws = (char*)d_ws; size_t off = 0;
  auto alloc = [&](size_t nbytes) -> void* {
    void* p = ws + off; off += (nbytes + 255) & ~(size_t)255; return p;
  };
  float* hu = (float*)alloc((size_t)NU * HIDF * 4);
  float* hi = (float*)alloc((size_t)NI * HIDF * 4);
  float* zA = (float*)alloc((size_t)NU * HIDF * 4);   // reused across node-type passes
  float* zB = (float*)alloc((size_t)NU * HIDF * 4);
  float* eal = (float*)alloc((size_t)NEDGE * NHEAD * 4);
  float* u_ad_iu = (float*)alloc((size_t)NU * NHEAD * 4);
  float* u_as_uu = (float*)alloc((size_t)NU * NHEAD * 4);
  float* u_ad_uu = (float*)alloc((size_t)NU * NHEAD * 4);
  float* u_as_ui = (float*)alloc((size_t)NU * NHEAD * 4);
  float* i_as_iu = (float*)alloc((size_t)NI * NHEAD * 4);
  float* i_ad_ui = (float*)alloc((size_t)NI * NHEAD * 4);
  float* i_as_ii = (float*)alloc((size_t)NI * NHEAD * 4);
  float* i_ad_ii = (float*)alloc((size_t)NI * NHEAD * 4);
  unsigned* amaxk = (unsigned*)alloc((size_t)NU * NHEAD * 4);
  float* denom = (float*)alloc((size_t)NU * NHEAD * 4);
  float* csA   = (float*)alloc(HIDF * 4);
  float* csB   = (float*)alloc(HIDF * 4);
  float* attnW = (float*)alloc(2 * 4);

  auto nblk = [](long long waves) { return (int)((waves + 7) / 8); };

  // 1) projections (WMMA f32)
  {
    long long waves = (long long)(NU / 16) * (HIDF / 64);
    wmma_gemm_bias<<<nblk(waves), 256, 0, stream>>>(x_user, Wp_user, bp_user, hu, NU, FIN, HIDF);
    wmma_gemm_bias<<<nblk(waves), 256, 0, stream>>>(x_item, Wp_item, bp_item, hi, NI, FIN, HIDF);
  }

  // 2) per-node attention logits
  node_logits<<<(NU * NHEAD + 255) / 256, 256, 0, stream>>>(
      hu, ad_iu, as_uu, ad_uu, as_ui, u_ad_iu, u_as_uu, u_ad_uu, u_as_ui, NU);
  node_logits<<<(NI * NHEAD + 255) / 256, 256, 0, stream>>>(
      hi, as_iu, ad_ui, as_ii, ad_ii, i_as_iu, i_ad_ui, i_as_ii, i_ad_ii, NI);

  // 3) edge relation pipeline
  auto run_rel = [&](const int* edge, const float* xs, const float* alsrc,
                     const float* aldst, int ndst, float* z) {
    const int* src = edge;
    const int* dst = edge + NEDGE;
    rel_init  <<<(ndst * HIDF + 255) / 256, 256, 0, stream>>>(amaxk, denom, z, ndst);
    rel_logits<<<(NEDGE * NHEAD + 255) / 256, 256, 0, stream>>>(src, dst, alsrc, aldst, eal, amaxk, NEDGE);
    rel_exp   <<<(NEDGE * NHEAD + 255) / 256, 256, 0, stream>>>(dst, eal, amaxk, denom, NEDGE);
    rel_agg   <<<(NEDGE + 3) / 4, 256, 0, stream>>>(src, dst, xs, eal, denom, z, NEDGE);
  };

  // 4) semantic attention + final linear
  auto run_group_final = [&](float* z0, float* z1, float* outp, int n) {
    zero_f32<<<1, 256, 0, stream>>>(csA, HIDF);
    zero_f32<<<1, 256, 0, stream>>>(csB, HIDF);
    long long waves = (long long)(n / 16) * (HIDF / 64);
    wmma_tanh_colsum<<<nblk(waves), 256, 0, stream>>>(z0, Wk, bk, csA, n);
    wmma_tanh_colsum<<<nblk(waves), 256, 0, stream>>>(z1, Wk, bk, csB, n);
    sem_softmax<<<1, 256, 0, stream>>>(csA, csB, q, attnW, 1.0f / (float)n);
    wmma_final<<<nblk(n / 16), 256, 0, stream>>>(z0, z1, attnW, Wl, bl, outp, n);
  };

  float* out_u = (float*)d_out;
  float* out_i = out_u + (size_t)NU * OUTF;

  // user destination pass: metapaths [iu, uu]
  run_rel(edge_iu, hi, i_as_iu, u_ad_iu, NU, zA);
  run_rel(edge_uu, hu, u_as_uu, u_ad_uu, NU, zB);
  run_group_final(zA, zB, out_u, NU);

  // item destination pass: metapaths [ui, ii]
  run_rel(edge_ui, hu, u_as_ui, i_ad_ui, NI, zA);
  run_rel(edge_ii, hi, i_as_ii, i_ad_ii, NI, zB);
  run_group_final(zA, zB, out_i, NI);
}